// GraphConv_40450001994130
// MI455X (gfx1250) — compile-verified
//
#include <hip/hip_runtime.h>

typedef __attribute__((ext_vector_type(2))) float v2f;
typedef __attribute__((ext_vector_type(8))) float v8f;

#define NNODES 50000
#define NEDGES 800000
#define NBATCH 2
#define NFEAT  64
#define NROWS  (NBATCH * NNODES)          // 100000 flat rows of xw
#define NTILES (NROWS / 16)               // 6250, exact
#define WSTRIDE 136                       // floats per K-pair row in LDS (8B-aligned pairs)

// ---------------- degree / normalization ----------------
__global__ void deg_init_kernel(float* __restrict__ deg, int n) {
    int i = blockIdx.x * blockDim.x + threadIdx.x;
    if (i < n) deg[i] = 1.0f;             // self-loop weight = 1
}

__global__ void deg_accum_kernel(const int* __restrict__ rows,
                                 const float* __restrict__ ew,
                                 float* __restrict__ deg, int e) {
    int i = blockIdx.x * blockDim.x + threadIdx.x;
    if (i < e) atomicAdd(&deg[rows[i]], ew[i]);
}

__global__ void dinv_kernel(float* __restrict__ deg, int n) {
    int i = blockIdx.x * blockDim.x + threadIdx.x;
    if (i < n) {
        float d = deg[i];
        deg[i] = (d > 0.0f) ? rsqrtf(d) : 0.0f;   // in place: deg -> dinv
    }
}

__global__ void lap_kernel(const int* __restrict__ rows,
                           const int* __restrict__ cols,
                           const float* __restrict__ ew,
                           const float* __restrict__ dinv,
                           float* __restrict__ lap, int e) {
    int i = blockIdx.x * blockDim.x + threadIdx.x;
    if (i < e) lap[i] = dinv[rows[i]] * ew[i] * dinv[cols[i]];
}

// ---------------- dense transform: xw = X (100000x64) * W (64x64) ----------------
// One wave computes one 16-row tile across all 64 output columns using
// V_WMMA_F32_16X16X4_F32 (4 accumulators, K-loop of 16 steps of 4).
// W is staged in LDS K-pair-interleaved: wl2[(k/2)*WSTRIDE + c*2 + (k&1)],
// so each B fragment {W[kb][c], W[kb+1][c]} is ONE aligned ds_load_b64.
__global__ void __launch_bounds__(256) gemm_wmma_kernel(const float* __restrict__ x,
                                                        const float* __restrict__ w,
                                                        float* __restrict__ xw,
                                                        int ntiles) {
    __shared__ float wl2[32 * WSTRIDE];
    int t = threadIdx.x;
    for (int i = t; i < 64 * 64; i += 256) {
        int k = i >> 6;                   // row of W (K index)
        int c = i & 63;                   // col of W (N index)
        wl2[(k >> 1) * WSTRIDE + c * 2 + (k & 1)] = w[i];
    }
    __syncthreads();

    int wave = t >> 5;
    int lane = t & 31;
    int tile = blockIdx.x * 8 + wave;
    if (tile >= ntiles) return;           // whole-wave guard: EXEC stays all-ones

    int m    = lane & 15;                 // A: row-in-tile / B,D: col-in-16
    int half = lane >> 4;                 // K sub-phase selector

    const float* xrow = x + (size_t)(tile * 16 + m) * NFEAT + 2 * half;

    v8f acc0 = {}, acc1 = {}, acc2 = {}, acc3 = {};

    #pragma unroll
    for (int k = 0; k < 64; k += 4) {
        // A fragment 16x4 f32: VGPR0 = K=k+2*half, VGPR1 = K=k+2*half+1
        v2f a = *(const v2f*)(xrow + k);
        int p = (k >> 1) + half;          // K-pair row index (kb = 2p, always even)
        const float* wb = &wl2[p * WSTRIDE + m * 2];
        v2f b0 = *(const v2f*)(wb +  0);  // cols m +  0
        v2f b1 = *(const v2f*)(wb + 32);  // cols m + 16
        v2f b2 = *(const v2f*)(wb + 64);  // cols m + 32
        v2f b3 = *(const v2f*)(wb + 96);  // cols m + 48
        acc0 = __builtin_amdgcn_wmma_f32_16x16x4_f32(false, a, false, b0, (short)0, acc0, false, false);
        acc1 = __builtin_amdgcn_wmma_f32_16x16x4_f32(false, a, false, b1, (short)0, acc1, false, false);
        acc2 = __builtin_amdgcn_wmma_f32_16x16x4_f32(false, a, false, b2, (short)0, acc2, false, false);
        acc3 = __builtin_amdgcn_wmma_f32_16x16x4_f32(false, a, false, b3, (short)0, acc3, false, false);
    }

    // D layout: VGPR v -> row (v + 8*half), col = m (+16 per accumulator)
    float* orow = xw + (size_t)tile * 16 * NFEAT;
    #pragma unroll
    for (int v = 0; v < 8; ++v) {
        int r = v + 8 * half;
        orow[r * NFEAT + m     ] = acc0[v];
        orow[r * NFEAT + m + 16] = acc1[v];
        orow[r * NFEAT + m + 32] = acc2[v];
        orow[r * NFEAT + m + 48] = acc3[v];
    }
}

// ---------------- out = bias + dinv[n]^2 * xw  (self-loop term) ----------------
__global__ void selfloop_bias_kernel(const float* __restrict__ xw,
                                     const float* __restrict__ dinv,
                                     const float* __restrict__ bias,
                                     float* __restrict__ out, int total) {
    int i = blockIdx.x * blockDim.x + threadIdx.x;
    if (i < total) {
        int f = i & 63;
        int n = (i >> 6) % NNODES;
        float di = dinv[n];
        out[i] = bias[f] + di * di * xw[i];
    }
}

// ---------------- edge scatter: out[b,row,f] += lap[e] * xw[b,col,f] ----------------
// 64 consecutive threads share one edge -> coalesced gathers and coalesced atomics.
__global__ void scatter_kernel(const int* __restrict__ rows,
                               const int* __restrict__ cols,
                               const float* __restrict__ lap,
                               const float* __restrict__ xw,
                               float* __restrict__ out, int e) {
    int t = blockIdx.x * blockDim.x + threadIdx.x;
    if (t >= e * NFEAT) return;
    int ee = t >> 6;
    int f  = t & 63;
    float l = lap[ee];
    int r = rows[ee];
    int c = cols[ee];
    float v0 = xw[(size_t)c * NFEAT + f];
    float v1 = xw[(size_t)(NNODES + c) * NFEAT + f];
    atomicAdd(&out[(size_t)r * NFEAT + f], l * v0);
    atomicAdd(&out[(size_t)(NNODES + r) * NFEAT + f], l * v1);
}

extern "C" void kernel_launch(void* const* d_in, const int* in_sizes, int n_in,
                              void* d_out, int out_size, void* d_ws, size_t ws_size,
                              hipStream_t stream) {
    const float* x    = (const float*)d_in[0];   // [B,N,64]
    const int*   ei   = (const int*)d_in[1];     // [2,E]
    const float* ew   = (const float*)d_in[2];   // [E]
    const float* w    = (const float*)d_in[3];   // [64,64]
    const float* bias = (const float*)d_in[4];   // [64]
    float* out = (float*)d_out;                  // [B,N,64]

    float* ws   = (float*)d_ws;
    float* deg  = ws;                            // N floats (becomes dinv in place)
    float* lap  = ws + NNODES;                   // E floats
    float* xw   = ws + NNODES + NEDGES;          // B*N*64 floats

    const int* rows = ei;
    const int* cols = ei + NEDGES;

    deg_init_kernel<<<(NNODES + 255) / 256, 256, 0, stream>>>(deg, NNODES);
    deg_accum_kernel<<<(NEDGES + 255) / 256, 256, 0, stream>>>(rows, ew, deg, NEDGES);
    dinv_kernel<<<(NNODES + 255) / 256, 256, 0, stream>>>(deg, NNODES);
    lap_kernel<<<(NEDGES + 255) / 256, 256, 0, stream>>>(rows, cols, ew, deg, lap, NEDGES);

    gemm_wmma_kernel<<<(NTILES + 7) / 8, 256, 0, stream>>>(x, w, xw, NTILES);

    int total = NBATCH * NNODES * NFEAT;
    selfloop_bias_kernel<<<(total + 255) / 256, 256, 0, stream>>>(xw, deg, bias, out, total);

    int sthreads = NEDGES * NFEAT;               // 51.2M
    scatter_kernel<<<(sthreads + 255) / 256, 256, 0, stream>>>(rows, cols, lap, xw, out, NEDGES);
}